// MockRenderer_22978075034254
// MI455X (gfx1250) — compile-verified
//
#include <hip/hip_runtime.h>
#include <stdint.h>

#define TILE     16
#define IMG_H    512
#define IMG_W    512
#define NTW      32
#define NTH      32
#define NT       (NTW * NTH)
#define K_CAP    256
#define CAPB     2048            // LDS candidate buffer (keys) per tile
#define PSX      (1.0f / 512.0f)
#define PSY      (1.0f / 512.0f)
#define XL_C     (-0.5f)
#define YT_C     (-0.5f)
#define TPS      (16.0f / 512.0f)
#define T_THRESH 1e-4f

typedef unsigned long long u64;
typedef __attribute__((ext_vector_type(4))) unsigned int u32x4;
typedef __attribute__((ext_vector_type(8))) unsigned int u32x8;

// ---------------------------------------------------------------------------
// TDM helper: 1-D DMA of `ndwords` dwords from global -> LDS (ndwords < 65536)
// D# per CDNA5 ISA: group0 {count=1, lds_addr,
// global_addr[56:0], type=2}; group1 {mask=0, data_size=4B, tensor_dim0=L,
// tensor_dim1=1, tile_dim0=L, tile_dim1=1, stride0=L}.
// ---------------------------------------------------------------------------
__device__ __forceinline__ void tdm_load_1d(unsigned lds_off, u64 gaddr, unsigned ndwords)
{
    unsigned lo  = (unsigned)__builtin_amdgcn_readfirstlane((int)lds_off);
    unsigned glo = (unsigned)__builtin_amdgcn_readfirstlane((int)(unsigned)gaddr);
    unsigned ghi = (unsigned)__builtin_amdgcn_readfirstlane((int)(unsigned)(gaddr >> 32));
    unsigned L   = (unsigned)__builtin_amdgcn_readfirstlane((int)ndwords);
    u32x4 g0;
    g0[0] = 1u;                                    // count=1 (valid descriptor)
    g0[1] = lo;                                    // lds_addr (bytes)
    g0[2] = glo;                                   // global_addr[31:0]
    g0[3] = (ghi & 0x01FFFFFFu) | 0x80000000u;     // global_addr[56:32] | type=2
    u32x8 g1;
    g1[0] = 0x00020000u;                           // data_size=2 (4 bytes), mask=0
    g1[1] = (L & 0xFFFFu) << 16;                   // tensor_dim0[15:0]
    g1[2] = (L >> 16) | (1u << 16);                // tensor_dim0[31:16] | tensor_dim1=1
    g1[3] = (L << 16);                             // tile_dim0 = L
    g1[4] = 1u;                                    // tile_dim1=1, tile_dim2=0
    g1[5] = L;                                     // tensor_dim0_stride lo32
    g1[6] = 0u;
    g1[7] = 0u;
    asm volatile("tensor_load_to_lds %0, %1" : : "s"(g0), "s"(g1) : "memory");
}

// ---------------------------------------------------------------------------
// Kernel 1: per-Gaussian projection / covariance / culling precompute
// ---------------------------------------------------------------------------
__global__ __launch_bounds__(256)
void gs_prep_kernel(const float* __restrict__ mean, const float* __restrict__ qvec,
                    const float* __restrict__ log_svec, const float* __restrict__ alpha,
                    const float* __restrict__ c2w,
                    float4* __restrict__ cull,              // {mx, my, r2d^2 (or -1), z}
                    float* __restrict__ c00o, float* __restrict__ c01o,
                    float* __restrict__ c11o, float* __restrict__ ao, int n)
{
    int i = blockIdx.x * 256 + threadIdx.x;
    if (i >= n) return;

    float R[3][3], tt[3];
#pragma unroll
    for (int r = 0; r < 3; r++) {
        R[r][0] = c2w[r*4+0]; R[r][1] = c2w[r*4+1]; R[r][2] = c2w[r*4+2];
        tt[r]   = c2w[r*4+3];
    }
    float d0 = mean[i*3+0] - tt[0], d1 = mean[i*3+1] - tt[1], d2 = mean[i*3+2] - tt[2];
    // p = (mean - t) @ R  (row vector times R)
    float p0 = d0*R[0][0] + d1*R[1][0] + d2*R[2][0];
    float p1 = d0*R[0][1] + d1*R[1][1] + d2*R[2][1];
    float p2 = d0*R[0][2] + d1*R[1][2] + d2*R[2][2];
    float z = p2, invz = 1.0f / z;
    float mx = p0 * invz, my = p1 * invz;

    float s0 = expf(log_svec[i*3+0]), s1 = expf(log_svec[i*3+1]), s2 = expf(log_svec[i*3+2]);

    float qw = qvec[i*4+0], qx = qvec[i*4+1], qy = qvec[i*4+2], qz = qvec[i*4+3];
    float qn = sqrtf(qw*qw + qx*qx + qy*qy + qz*qz);
    float iq = 1.0f / qn; qw *= iq; qx *= iq; qy *= iq; qz *= iq;

    float Rg[3][3];
    Rg[0][0] = 1.0f - 2.0f*(qy*qy + qz*qz); Rg[0][1] = 2.0f*(qx*qy - qw*qz); Rg[0][2] = 2.0f*(qx*qz + qw*qy);
    Rg[1][0] = 2.0f*(qx*qy + qw*qz); Rg[1][1] = 1.0f - 2.0f*(qx*qx + qz*qz); Rg[1][2] = 2.0f*(qy*qz - qw*qx);
    Rg[2][0] = 2.0f*(qx*qz - qw*qy); Rg[2][1] = 2.0f*(qy*qz + qw*qx); Rg[2][2] = 1.0f - 2.0f*(qx*qx + qy*qy);

    // M = Rg * diag(svec) ; Sigma = M M^T
    float M[3][3];
#pragma unroll
    for (int a = 0; a < 3; a++) { M[a][0] = Rg[a][0]*s0; M[a][1] = Rg[a][1]*s1; M[a][2] = Rg[a][2]*s2; }
    float Sig[3][3];
#pragma unroll
    for (int a = 0; a < 3; a++)
#pragma unroll
        for (int b = 0; b < 3; b++)
            Sig[a][b] = M[a][0]*M[b][0] + M[a][1]*M[b][1] + M[a][2]*M[b][2];

    // J = [[invz,0,-p0*invz^2],[0,invz,-p1*invz^2]];  JW = J R^T (2x3)
    float J02 = -p0*invz*invz, J12 = -p1*invz*invz;
    float JW0[3], JW1[3];
#pragma unroll
    for (int b = 0; b < 3; b++) { JW0[b] = invz*R[b][0] + J02*R[b][2]; JW1[b] = invz*R[b][1] + J12*R[b][2]; }

    float T0[3], T1[3];
#pragma unroll
    for (int j = 0; j < 3; j++) {
        T0[j] = JW0[0]*Sig[0][j] + JW0[1]*Sig[1][j] + JW0[2]*Sig[2][j];
        T1[j] = JW1[0]*Sig[0][j] + JW1[1]*Sig[1][j] + JW1[2]*Sig[2][j];
    }
    float cov00 = T0[0]*JW0[0] + T0[1]*JW0[1] + T0[2]*JW0[2];
    float cov01 = T0[0]*JW1[0] + T0[1]*JW1[1] + T0[2]*JW1[2];
    float cov10 = T1[0]*JW0[0] + T1[1]*JW0[1] + T1[2]*JW0[2];
    float cov11 = T1[0]*JW1[0] + T1[1]*JW1[1] + T1[2]*JW1[2];
    float c01s = 0.5f * (cov01 + cov10);

    float r3 = 3.0f * fmaxf(s0, fmaxf(s1, s2));
    float nf = sqrtf(1.25f);
    float a1 = 1.0f / nf, b1 = 0.5f / nf;
    bool inside = (( p0*a1 + p2*b1) > -r3) && ((-p0*a1 + p2*b1) > -r3) &&
                  (( p1*a1 + p2*b1) > -r3) && ((-p1*a1 + p2*b1) > -r3) &&
                  (z > 0.1f - r3) && (z < 100.0f + r3);
    bool valid = inside && (z > 1e-6f);

    float mm   = 0.5f * (cov00 + cov11);
    float pdet = cov00*cov11 - c01s*c01s;
    float disc = mm*mm - pdet; disc = (disc < 0.0f) ? 0.0f : disc;     // NaN-preserving clip
    float radius = sqrtf(mm + sqrtf(disc));
    float r2d = radius * 3.0f;
    float det_c = (pdet < 1e-12f) ? 1e-12f : pdet;                     // NaN-preserving clip

    c00o[i] =  cov11 / det_c;
    c01o[i] = (-c01s) / det_c;
    c11o[i] =  cov00 / det_c;
    ao[i]   = 1.0f / (1.0f + expf(-alpha[i]));
    cull[i] = make_float4(mx, my, valid ? r2d * r2d : -1.0f, z);
}

// ---------------------------------------------------------------------------
// Kernel 2: per-tile top-K (by z, tie-break by index) via LDS buffer + bitonic,
// with cull data streamed through LDS by double-buffered TDM DMA.
// ---------------------------------------------------------------------------
__device__ __forceinline__ void bitonic_sort_2048(u64* buf, int tid)
{
    for (unsigned k = 2; k <= CAPB; k <<= 1) {
        for (unsigned j = k >> 1; j > 0; j >>= 1) {
            for (unsigned idx = tid; idx < CAPB; idx += 256) {
                unsigned ixj = idx ^ j;
                if (ixj > idx) {
                    u64 a = buf[idx], b = buf[ixj];
                    bool up = ((idx & k) == 0);
                    if ((a > b) == up) { buf[idx] = b; buf[ixj] = a; }
                }
            }
            __syncthreads();
        }
    }
}

__device__ __forceinline__ void topk_prune(u64* buf, int* cnt, u64* thr, int tid)
{
    int c = *cnt;
    for (int idx = tid; idx < CAPB; idx += 256)
        if (idx >= c) buf[idx] = ~0ULL;
    __syncthreads();
    bitonic_sort_2048(buf, tid);     // ends with __syncthreads
    if (tid == 0) { *cnt = (c < K_CAP) ? c : K_CAP; *thr = buf[K_CAP - 1]; }
    __syncthreads();
}

__global__ __launch_bounds__(256)
void gs_bin_kernel(const float4* __restrict__ cull,
                   const float* __restrict__ c00, const float* __restrict__ c01,
                   const float* __restrict__ c11, const float* __restrict__ aa,
                   const float* __restrict__ color,
                   float* __restrict__ tileParams, int* __restrict__ tileCount, int n)
{
    __shared__ u64 buf[CAPB];
    __shared__ float4 stage[2][256];               // double-buffered TDM staging (2 x 4KB)
    __shared__ int cnt;
    __shared__ u64 thr;
    const int t = blockIdx.x, tid = threadIdx.x;
    const int ti = t / NTW, tj = t % NTW;
    const float tx0 = XL_C + tj * TPS, tx1 = tx0 + TPS;
    const float ty0 = YT_C + ti * TPS, ty1 = ty0 + TPS;

    const unsigned st0 = (unsigned)(size_t)(void*)&stage[0][0];
    const unsigned st1 = (unsigned)(size_t)(void*)&stage[1][0];
    const int rounds = (n + 255) / 256;

    if (tid == 0) { cnt = 0; thr = ~0ULL; }
    if (tid < 32) {                                // wave 0: prefetch chunk 0
        int len0 = (n < 256) ? n : 256;
        tdm_load_1d(st0, (u64)(size_t)cull, (unsigned)len0 * 4u);
    }
    __syncthreads();

    for (int r = 0; r < rounds; r++) {
        const int base = r * 256;
        if (tid < 32) {                            // wave 0: prefetch next, wait current
            if (r + 1 < rounds) {
                int nb  = base + 256;
                int len = n - nb; if (len > 256) len = 256;
                tdm_load_1d(((r + 1) & 1) ? st1 : st0,
                            (u64)(size_t)(cull + nb), (unsigned)len * 4u);
                __builtin_amdgcn_s_wait_tensorcnt(1);   // chunk r landed (TDM in-order)
            } else {
                __builtin_amdgcn_s_wait_tensorcnt(0);
            }
        }
        __syncthreads();                           // stage[r&1] visible to all waves

        int g = base + tid;
        bool ins = false; u64 key = 0;
        if (g < n) {
            float4 cu = stage[r & 1][tid];
            float dx = fmaxf(fmaxf(tx0 - cu.x, cu.x - tx1), 0.0f);
            float dy = fmaxf(fmaxf(ty0 - cu.y, cu.y - ty1), 0.0f);
            if (dx*dx + dy*dy <= cu.z) {           // cu.z = r2d^2 (or -1 / NaN)
                key = ((u64)__float_as_uint(cu.w) << 32) | (unsigned)g;
                ins = true;
            }
        }
        int c = cnt;                               // uniform: no writes since last barrier
        __syncthreads();                           // everyone snapshotted cnt
        if (c + 256 > CAPB) topk_prune(buf, &cnt, &thr, tid);   // uniform
        if (ins && key < thr) { int pos = atomicAdd(&cnt, 1); buf[pos] = key; }
        __syncthreads();                           // end of round: stage reads + inserts done
    }
    topk_prune(buf, &cnt, &thr, tid);              // final sort ascending (z, idx)

    int c = cnt;
    if (tid == 0) tileCount[t] = c;
    if (tid < c) {
        unsigned id = (unsigned)(buf[tid] & 0xFFFFFFFFu);
        float4 cu = cull[id];
        float* dst = tileParams + (size_t)t * (K_CAP * 9) + tid * 9;
        dst[0] = cu.x;   dst[1] = cu.y;
        dst[2] = c00[id]; dst[3] = c01[id]; dst[4] = c11[id];
        dst[5] = color[id*3+0]; dst[6] = color[id*3+1]; dst[7] = color[id*3+2];
        dst[8] = aa[id];
    }
}

// ---------------------------------------------------------------------------
// Kernel 3: per-tile alpha blending; params staged to LDS via TDM
// ---------------------------------------------------------------------------
__global__ __launch_bounds__(256)
void gs_render_kernel(const float* __restrict__ tileParams,
                      const int* __restrict__ tileCount, float* __restrict__ out)
{
    __shared__ float sp[K_CAP * 9];                // 9216 B staged by TDM
    const int t = blockIdx.x, tid = threadIdx.x;
    const int ti = t / NTW, tj = t % NTW;

    int cnt = tileCount[t];
    cnt = __builtin_amdgcn_readfirstlane(cnt);

    if (cnt > 0) {
        if (tid < 32) {                            // wave 0 issues the TDM
            tdm_load_1d((unsigned)(size_t)(void*)sp,
                        (u64)(size_t)(tileParams + (size_t)t * (K_CAP * 9)),
                        (unsigned)cnt * 9u);
            __builtin_amdgcn_s_wait_tensorcnt(0);
        }
    }
    __syncthreads();

    const int px = tid & 15, py = tid >> 4;
    const float X = XL_C + ((float)(tj * TILE + px) + 0.5f) * PSX;
    const float Y = YT_C + ((float)(ti * TILE + py) + 0.5f) * PSY;

    float C0 = 0.0f, C1 = 0.0f, C2 = 0.0f, T = 1.0f;
    for (int k = 0; k < cnt; k++) {
        const float* gp = &sp[k * 9];
        float mu0 = gp[0], mu1 = gp[1];
        float ci00 = gp[2], ci01 = gp[3], ci11 = gp[4];
        float co0 = gp[5], co1 = gp[6], co2 = gp[7], al = gp[8];
        if (T > T_THRESH) {
            float dX = X - mu0, dY = Y - mu1;
            float pw = -0.5f * (ci00*dX*dX + 2.0f*ci01*dX*dY + ci11*dY*dY);
            float w = al * expf(pw);
            w = (w > 0.99f) ? 0.99f : w;           // NaN-preserving min
            float tw = T * w;
            C0 += tw * co0; C1 += tw * co1; C2 += tw * co2;
            T = T * (1.0f - w);
        }
        if ((k & 31) == 31) {                      // uniform early-out, result-invariant
            if (__syncthreads_count(T > T_THRESH) == 0) break;
        }
    }
    int y = ti * TILE + py, x = tj * TILE + px;
    size_t o = ((size_t)y * IMG_W + x) * 3;
    out[o + 0] = C0; out[o + 1] = C1; out[o + 2] = C2;
}

// ---------------------------------------------------------------------------
extern "C" void kernel_launch(void* const* d_in, const int* in_sizes, int n_in,
                              void* d_out, int out_size, void* d_ws, size_t ws_size,
                              hipStream_t stream)
{
    const float* mean     = (const float*)d_in[0];
    const float* qvec     = (const float*)d_in[1];
    const float* log_svec = (const float*)d_in[2];
    const float* color    = (const float*)d_in[3];
    const float* alpha    = (const float*)d_in[4];
    const float* c2w      = (const float*)d_in[5];
    const int n = in_sizes[4];

    char* ws = (char*)d_ws;
    float4* cull = (float4*)ws;            ws += (size_t)n * sizeof(float4);
    float* c00   = (float*)ws;             ws += (size_t)n * sizeof(float);
    float* c01   = (float*)ws;             ws += (size_t)n * sizeof(float);
    float* c11   = (float*)ws;             ws += (size_t)n * sizeof(float);
    float* aa    = (float*)ws;             ws += (size_t)n * sizeof(float);
    int* tileCount = (int*)ws;             ws += (size_t)NT * sizeof(int);
    ws = (char*)(((size_t)ws + 15) & ~(size_t)15);
    float* tileParams = (float*)ws;        // NT * 256 * 9 floats (~9.4 MB)

    gs_prep_kernel<<<(n + 255) / 256, 256, 0, stream>>>(
        mean, qvec, log_svec, alpha, c2w, cull, c00, c01, c11, aa, n);

    gs_bin_kernel<<<NT, 256, 0, stream>>>(
        cull, c00, c01, c11, aa, color, tileParams, tileCount, n);

    gs_render_kernel<<<NT, 256, 0, stream>>>(
        tileParams, tileCount, (float*)d_out);
}